// MsDroid_50775103373746
// MI455X (gfx1250) — compile-verified
//
#include <hip/hip_runtime.h>

// GIN forward for MI455X (gfx1250, wave32).
// GEMMs use V_WMMA_F32_16X16X4_F32 (exact fp32): workload is memory/atomic
// bound (~1.5-2 GB traffic @ 23.3 TB/s dominates 19.7 GFLOP fp32), so full
// precision WMMA is the right choice — the matrix pipe is not the bottleneck.
// Round 3: fix async-to-LDS builtin pointer types: it takes
// (int4 addrspace(1)*, int4 addrspace(3)*, imm offset, imm cpol).

#define N_NODES 100000
#define N_EDGES 600000
#define HDIM    128
#define NGRAPH  256

typedef float v2f __attribute__((ext_vector_type(2)));
typedef float v8f __attribute__((ext_vector_type(8)));

#if __has_builtin(__builtin_amdgcn_global_load_async_to_lds_b128)
#define USE_ASYNC_LDS 1
typedef int b128_t __attribute__((vector_size(16)));
typedef __attribute__((address_space(1))) b128_t* gptr_b128;
typedef __attribute__((address_space(3))) b128_t* lptr_b128;
#else
#define USE_ASYNC_LDS 0
#endif

__device__ __forceinline__ void wait_async_zero() {
#if USE_ASYNC_LDS
#if __has_builtin(__builtin_amdgcn_s_wait_asynccnt)
  __builtin_amdgcn_s_wait_asynccnt(0);
#else
  asm volatile("s_wait_asynccnt 0x0" ::: "memory");
#endif
#endif
}

// ---------------------------------------------------------------- copy (agg = h)
__global__ __launch_bounds__(256) void k_copy(const float* __restrict__ src,
                                              float* __restrict__ dst, int n4) {
  int t = blockIdx.x * 256 + threadIdx.x;
  if (t < n4) ((float4*)dst)[t] = ((const float4*)src)[t];
}

// ------------------------------------------- 128x128 transpose: Wt[n][k] = W[k][n]
__global__ __launch_bounds__(256) void k_transpose128(const float* __restrict__ W,
                                                      float* __restrict__ Wt) {
  int t = blockIdx.x * 256 + threadIdx.x;   // 16384 elements
  int k = t >> 7, n = t & 127;
  Wt[(size_t)n * HDIM + k] = W[(size_t)k * HDIM + n];
}

// ------------------------------------------------- scatter-add: agg[dst] += h[src]
__global__ __launch_bounds__(256) void k_scatter(const float* __restrict__ h,
                                                 const int* __restrict__ srcIdx,
                                                 const int* __restrict__ dstIdx,
                                                 float* __restrict__ agg) {
  int t = blockIdx.x * 256 + threadIdx.x;
  int e = t >> 5;                 // 32 lanes per edge, 4 floats per lane
  if (e >= N_EDGES) return;
  int l = t & 31;
  int s = srcIdx[e], d = dstIdx[e];
  float4 v = ((const float4*)(h + (size_t)s * HDIM))[l];
  float* o = agg + (size_t)d * HDIM + (size_t)l * 4;
  atomicAdd(o + 0, v.x);
  atomicAdd(o + 1, v.y);
  atomicAdd(o + 2, v.z);
  atomicAdd(o + 3, v.w);
}

// ---------------- fused GEMM: Y = epilogue(A[16-row tile] @ Wt^T + bias)
// Wt is the TRANSPOSED weight (Wt[n][k] = W[k][n]); epilogue = ReLU, then
// optional LayerNorm(gamma,beta) over the 128 features.
// Block = 256 threads = 8 waves; wave w computes cols [16w,16w+16).
__global__ __launch_bounds__(256) void k_gemm(const float* __restrict__ A,
                                              const float* __restrict__ Wt,
                                              const float* __restrict__ bias,
                                              const float* __restrict__ gamma,
                                              const float* __restrict__ beta,
                                              float* __restrict__ Y,
                                              int do_ln) {
  __shared__ float sm[16][132];          // A tile, reused as output tile (132%64=4 banks: conflict-free)
  __shared__ float mu_s[16], rs_s[16];

  const int tid  = threadIdx.x;
  const int wave = tid >> 5;
  const int lane = tid & 31;
  const int hi   = lane >> 4;            // lane group: 0 -> K{0,1}, 1 -> K{2,3}
  const int lm   = lane & 15;
  const size_t rowBase = (size_t)blockIdx.x * 16;

  // Stage the 16x128 A tile into LDS (2 x 16B per thread, no divergence).
  for (int i = tid; i < 16 * 32; i += 256) {
    int r = i >> 5, q = i & 31;
    const float* gp = A + (rowBase + (size_t)r) * HDIM + (size_t)q * 4;
#if USE_ASYNC_LDS
    __builtin_amdgcn_global_load_async_to_lds_b128(
        (gptr_b128)gp, (lptr_b128)&sm[r][q * 4], 0, 0);
#else
    *(float4*)&sm[r][q * 4] = *(const float4*)gp;
#endif
  }
  wait_async_zero();
  __syncthreads();

  const int col = wave * 16 + lm;        // this lane's output column
  v8f c = {};
  // K loop: 32 x v_wmma_f32_16x16x4_f32 (fully unrolled).
  // A frag (16x4 fp32): lanes 0-15 hold M=lm; VGPR0=K0,VGPR1=K1 (hi=0) / K2,K3 (hi=1)
  // B frag (4x16 fp32): mirrored — lane holds column N=lm, same K striping.
  //   b = {W[k][col], W[k+1][col]} = contiguous pair in transposed Wt[col][k..k+1].
#pragma unroll
  for (int k0 = 0; k0 < HDIM; k0 += 4) {
    v2f a = *(const v2f*)&sm[lm][k0 + 2 * hi];                         // ds_load_b64
    v2f b = *(const v2f*)&Wt[(size_t)col * HDIM + (k0 + 2 * hi)];      // global_load_b64
    c = __builtin_amdgcn_wmma_f32_16x16x4_f32(false, a, false, b, (short)0, c,
                                              false, false);
  }
  __syncthreads();                        // all waves done reading A tile

  // Bias + ReLU into LDS tile. C/D layout: VGPR i -> row i (lanes 0-15) / i+8 (16-31).
  float bc = bias[col];
#pragma unroll
  for (int i = 0; i < 8; i++) {
    float v = c[i] + bc;
    sm[i + 8 * hi][col] = fmaxf(v, 0.0f);
  }
  __syncthreads();

  if (do_ln) {
    if (tid < 16) {                       // one thread per row: mean & var
      float s = 0.0f, s2 = 0.0f;
#pragma unroll 8
      for (int j = 0; j < HDIM; j++) {
        float v = sm[tid][j];
        s += v; s2 += v * v;
      }
      float mu  = s * (1.0f / HDIM);
      float var = s2 * (1.0f / HDIM) - mu * mu;   // population var, matches jnp.var
      mu_s[tid] = mu;
      rs_s[tid] = rsqrtf(var + 1e-5f);
    }
    __syncthreads();
    int r = tid >> 4, c0 = tid & 15;
    float mu = mu_s[r], rs = rs_s[r];
    float* yr = Y + (rowBase + (size_t)r) * HDIM;
#pragma unroll
    for (int j = c0; j < HDIM; j += 16)
      yr[j] = (sm[r][j] - mu) * rs * gamma[j] + beta[j];
  } else {
    int r = tid >> 4, c0 = tid & 15;
    float* yr = Y + (rowBase + (size_t)r) * HDIM;
#pragma unroll
    for (int j = c0; j < HDIM; j += 16)
      yr[j] = sm[r][j];
  }
}

// ---------------------------------------------------- mean-pool accumulation
__global__ __launch_bounds__(256) void k_pool(const float* __restrict__ y,
                                              const int* __restrict__ batch,
                                              float* __restrict__ sums,
                                              float* __restrict__ cnts) {
  int t = blockIdx.x * 256 + threadIdx.x;
  int r = t >> 5;
  if (r >= N_NODES) return;
  int l = t & 31;
  int g = batch[r];
  float4 v = ((const float4*)(y + (size_t)r * HDIM))[l];
  float* o = sums + (size_t)g * HDIM + (size_t)l * 4;
  atomicAdd(o + 0, v.x);
  atomicAdd(o + 1, v.y);
  atomicAdd(o + 2, v.z);
  atomicAdd(o + 3, v.w);
  if (l == 0) atomicAdd(&cnts[g], 1.0f);
}

// ------------------------------------------- head: pooled @ Wp1 @ Wp2, log_softmax
__global__ __launch_bounds__(NGRAPH) void k_head(const float* __restrict__ sums,
                                                 const float* __restrict__ cnts,
                                                 const float* __restrict__ Wp1,
                                                 const float* __restrict__ bp1,
                                                 const float* __restrict__ Wp2,
                                                 const float* __restrict__ bp2,
                                                 float* __restrict__ out) {
  int g = threadIdx.x;                    // one graph per thread
  float inv = 1.0f / fmaxf(cnts[g], 1.0f);
  float pooled[HDIM];
#pragma unroll
  for (int k = 0; k < HDIM; k++) pooled[k] = sums[(size_t)g * HDIM + k] * inv;

  float e0 = bp2[0], e1 = bp2[1];
  for (int j = 0; j < HDIM; j++) {
    float pj = bp1[j];
#pragma unroll
    for (int k = 0; k < HDIM; k++) pj += pooled[k] * Wp1[(size_t)k * HDIM + j];
    e0 += pj * Wp2[j * 2 + 0];
    e1 += pj * Wp2[j * 2 + 1];
  }
  out[g * 2 + 0] = e0;
  out[g * 2 + 1] = e1;
  float m   = fmaxf(e0, e1);
  float lse = m + logf(expf(e0 - m) + expf(e1 - m));
  out[2 * NGRAPH + g * 2 + 0] = e0 - lse;
  out[2 * NGRAPH + g * 2 + 1] = e1 - lse;
}

extern "C" void kernel_launch(void* const* d_in, const int* in_sizes, int n_in,
                              void* d_out, int out_size, void* d_ws, size_t ws_size,
                              hipStream_t stream) {
  (void)in_sizes; (void)n_in; (void)out_size; (void)ws_size;

  const float* x     = (const float*)d_in[0];
  const int*   ei    = (const int*)d_in[1];
  const int*   src   = ei;                 // edge_index[0]
  const int*   dst   = ei + N_EDGES;       // edge_index[1]
  const int*   batch = (const int*)d_in[2];
  const float* W1[3] = {(const float*)d_in[3], (const float*)d_in[7],  (const float*)d_in[11]};
  const float* b1[3] = {(const float*)d_in[4], (const float*)d_in[8],  (const float*)d_in[12]};
  const float* W2[3] = {(const float*)d_in[5], (const float*)d_in[9],  (const float*)d_in[13]};
  const float* b2[3] = {(const float*)d_in[6], (const float*)d_in[10], (const float*)d_in[14]};
  const float* lng[2] = {(const float*)d_in[15], (const float*)d_in[17]};
  const float* lnb[2] = {(const float*)d_in[16], (const float*)d_in[18]};
  const float* Wp1 = (const float*)d_in[19];
  const float* bp1 = (const float*)d_in[20];
  const float* Wp2 = (const float*)d_in[21];
  const float* bp2 = (const float*)d_in[22];

  float* bufA = (float*)d_ws;                         // agg / GEMM1 input
  float* bufB = bufA + (size_t)N_NODES * HDIM;        // MLP hidden
  float* bufC = bufB + (size_t)N_NODES * HDIM;        // layer output
  float* sums = bufC + (size_t)N_NODES * HDIM;        // [G,128]
  float* cnts = sums + (size_t)NGRAPH * HDIM;         // [G] (contiguous after sums)
  float* wt   = cnts + NGRAPH;                        // 6 transposed 128x128 weights
  float* Wt1[3] = {wt,                  wt + 2 * HDIM * HDIM, wt + 4 * HDIM * HDIM};
  float* Wt2[3] = {wt + HDIM * HDIM,    wt + 3 * HDIM * HDIM, wt + 5 * HDIM * HDIM};

  const int n4       = N_NODES * HDIM / 4;
  const int copyGrid = (n4 + 255) / 256;
  const int scatGrid = (N_EDGES * 32 + 255) / 256;
  const int gemmGrid = N_NODES / 16;                  // 6250, exact
  const int poolGrid = (N_NODES * 32 + 255) / 256;
  const int trGrid   = (HDIM * HDIM) / 256;           // 64
  dim3 blk(256);

  // Pre-transpose all six MLP weights (64 KB each; amortized over 6250 blocks).
  for (int L = 0; L < 3; L++) {
    k_transpose128<<<trGrid, blk, 0, stream>>>(W1[L], Wt1[L]);
    k_transpose128<<<trGrid, blk, 0, stream>>>(W2[L], Wt2[L]);
  }

  const float* hcur = x;
  for (int L = 0; L < 3; L++) {
    k_copy   <<<copyGrid, blk, 0, stream>>>(hcur, bufA, n4);
    k_scatter<<<scatGrid, blk, 0, stream>>>(hcur, src, dst, bufA);
    k_gemm   <<<gemmGrid, blk, 0, stream>>>(bufA, Wt1[L], b1[L], nullptr, nullptr, bufB, 0);
    int do_ln = (L < 2) ? 1 : 0;
    k_gemm   <<<gemmGrid, blk, 0, stream>>>(bufB, Wt2[L], b2[L],
                                            do_ln ? lng[L] : nullptr,
                                            do_ln ? lnb[L] : nullptr, bufC, do_ln);
    hcur = bufC;
  }

  (void)hipMemsetAsync(sums, 0, sizeof(float) * ((size_t)NGRAPH * HDIM + NGRAPH), stream);
  k_pool<<<poolGrid, blk, 0, stream>>>(bufC, batch, sums, cnts);
  k_head<<<1, dim3(NGRAPH), 0, stream>>>(sums, cnts, Wp1, bp1, Wp2, bp2, (float*)d_out);
}